// SparseLayer_8194797601572
// MI455X (gfx1250) — compile-verified
//
#include <hip/hip_runtime.h>

// LCA sparse coding solver for MI455X (gfx1250), fp32 WMMA path.
// Round 2: B-operands (phi, Gram) pre-packed into WMMA-fragment order so each
// lane's fragment is one aligned float2 load (halves VMEM instr in hot loop).
//
// Problem sizes (fixed by the reference):
#define BATCH    8192
#define WH       4096      // 64*64 pixels per image
#define NF       256       // filters
#define SHRINK   0.25f
#define LRATE    0.1f
#define MAXITER  500
#define TOLSQ    (0.01f * 0.01f)   // compare ||du||^2 against TOL^2

typedef __attribute__((ext_vector_type(2))) float v2f;
typedef __attribute__((ext_vector_type(8))) float v8f;

__device__ __forceinline__ float thr(float x) { return x > SHRINK ? x : 0.0f; }

// ---------------------------------------------------------------------------
// init: zero u (iteration 0 input), the per-iteration norm accumulators, and
// the "current buffer" flag. Must run every launch (ws is not re-poisoned,
// and norms are accumulated with atomics).
// ---------------------------------------------------------------------------
__global__ void lca_init_kernel(float* __restrict__ uA,
                                float* __restrict__ norms,
                                int* __restrict__ cur) {
    const int idx = blockIdx.x * blockDim.x + threadIdx.x;
    if (idx < BATCH * NF) uA[idx] = 0.0f;
    if (idx < MAXITER + 8) norms[idx] = 0.0f;
    if (idx == 0) *cur = 0;
}

// ---------------------------------------------------------------------------
// pack phi [WH, NF] row-major -> K-pair interleaved fragment layout:
//   phiPack[(k>>1)*2*NF + n*2 + (k&1)] = phi[k*NF + n]
// A lane's WMMA B fragment (rows k, k+1 at column n) is then ONE float2 load.
// ---------------------------------------------------------------------------
__global__ void lca_pack_phi_kernel(const float* __restrict__ phi,
                                    float* __restrict__ phiPack) {
    const int idx = blockIdx.x * blockDim.x + threadIdx.x;   // < WH*NF
    const int k = idx >> 8;          // row (pixel)
    const int n = idx & (NF - 1);    // column (filter)
    phiPack[(size_t)(k >> 1) * (2 * NF) + n * 2 + (k & 1)] = phi[idx];
}

// ---------------------------------------------------------------------------
// Gram: G = phi^T phi, diagonal zeroed, written DIRECTLY in packed layout.
// One wave per 16x16 output tile (256 tiles), K=4096 in steps of 4 via
// V_WMMA_F32_16X16X4_F32. Both A and B fragments are single float2 loads
// from phiPack.
// ---------------------------------------------------------------------------
__global__ void lca_gram_kernel(const float* __restrict__ phiPackF,
                                float* __restrict__ Gpack) {
    const int lane = threadIdx.x & 31;
    const int wave = threadIdx.x >> 5;
    const int half = lane >> 4;      // 0: lanes 0-15, 1: lanes 16-31
    const int l16  = lane & 15;
    const int job  = blockIdx.x * 8 + wave;       // 0..255
    const int m0   = (job >> 4) * 16;
    const int n0   = (job & 15) * 16;

    const v2f* Phi2 = (const v2f*)phiPackF;       // [WH/2][NF] of float2

    v8f acc = {0.f,0.f,0.f,0.f,0.f,0.f,0.f,0.f};

    #pragma unroll 4
    for (int k0 = 0; k0 < WH; k0 += 4) {
        const v2f* row = Phi2 + (size_t)((k0 >> 1) + half) * NF;
        v2f A  = row[m0 + l16];   // (phi[k0+2h][m], phi[k0+2h+1][m])
        v2f Bv = row[n0 + l16];   // (phi[k0+2h][n], phi[k0+2h+1][n])
        acc = __builtin_amdgcn_wmma_f32_16x16x4_f32(
            false, A, false, Bv, (short)0, acc, false, false);
    }

    #pragma unroll
    for (int r = 0; r < 8; ++r) {
        const int row = m0 + r + 8 * half;
        const int col = n0 + l16;
        const float v = (row == col) ? 0.0f : acc[r];     // zero diagonal
        Gpack[(size_t)(row >> 1) * (2 * NF) + col * 2 + (row & 1)] = v;
    }
}

// ---------------------------------------------------------------------------
// excite = images @ phi.  images is [BATCH, WH] row-major.
// One wave computes a 16x64 strip (4 accumulators share the A fragment).
// 512 M-tiles x 4 N-groups = 2048 wave-jobs -> 256 blocks of 8 waves.
// ---------------------------------------------------------------------------
__global__ void lca_excite_kernel(const float* __restrict__ images,
                                  const float* __restrict__ phiPackF,
                                  float* __restrict__ excite) {
    const int lane = threadIdx.x & 31;
    const int wave = threadIdx.x >> 5;
    const int half = lane >> 4;
    const int l16  = lane & 15;
    const int job  = blockIdx.x * 8 + wave;       // 0..2047
    const int m0   = (job >> 2) * 16;             // 512 M tiles
    const int nB   = (job & 3) * 64;              // 4 N-tiles of 16

    const v2f* Phi2 = (const v2f*)phiPackF;

    v8f zero8 = {0.f,0.f,0.f,0.f,0.f,0.f,0.f,0.f};
    v8f acc[4] = {zero8, zero8, zero8, zero8};

    // A fragment: row (m0+l16) of images, K pair (k0+2*half, +1) -> aligned float2.
    const float* arow = images + (size_t)(m0 + l16) * WH + 2 * half;

    #pragma unroll 2
    for (int k0 = 0; k0 < WH; k0 += 4) {
        v2f A = *(const v2f*)(arow + k0);
        const v2f* bRow = Phi2 + (size_t)((k0 >> 1) + half) * NF + l16;
        #pragma unroll
        for (int t = 0; t < 4; ++t) {
            v2f Bv = bRow[nB + t * 16];
            acc[t] = __builtin_amdgcn_wmma_f32_16x16x4_f32(
                false, A, false, Bv, (short)0, acc[t], false, false);
        }
    }

    #pragma unroll
    for (int t = 0; t < 4; ++t) {
        const int col = nB + t * 16 + l16;
        #pragma unroll
        for (int r = 0; r < 8; ++r)
            excite[(m0 + r + 8 * half) * NF + col] = acc[t][r];
    }
}

// ---------------------------------------------------------------------------
// One LCA iteration:
//   act = threshold(u_in); du = excite - u_in - act @ G; u_out = u_in + LR*du
//   norms[iter] += ||du||^2 (fp32 atomics, wave-reduced first)
// Early-exits (and cascades the exit) when the previous iteration's ||du||^2
// fell below TOL^2. Ping-pong buffers avoid the read/write race on u.
// Hot loop: 1 float2 (u/A) + 4 float2 (packed G/B, one clause) per 4 WMMAs.
// ---------------------------------------------------------------------------
__global__ void lca_iter_kernel(const float* __restrict__ u_in,
                                float* __restrict__ u_out,
                                const float* __restrict__ excite,
                                const float* __restrict__ GpackF,
                                float* __restrict__ norms,
                                int* __restrict__ cur,
                                int iter, int out_is_b) {
    if (iter > 0 && norms[iter - 1] < TOLSQ) return;   // converged: skip (cascades)

    const int lane = threadIdx.x & 31;
    const int wave = threadIdx.x >> 5;
    const int half = lane >> 4;
    const int l16  = lane & 15;
    const int job  = blockIdx.x * 8 + wave;       // 0..2047
    const int m0   = (job >> 2) * 16;
    const int nB   = (job & 3) * 64;

    const v2f* G2 = (const v2f*)GpackF;           // [NF/2][NF] of float2

    v8f zero8 = {0.f,0.f,0.f,0.f,0.f,0.f,0.f,0.f};
    v8f acc[4] = {zero8, zero8, zero8, zero8};

    const float* urow = u_in + (size_t)(m0 + l16) * NF + 2 * half;

    #pragma unroll 2
    for (int k0 = 0; k0 < NF; k0 += 4) {
        v2f uv = *(const v2f*)(urow + k0);
        v2f A;  A.x = thr(uv.x);  A.y = thr(uv.y);          // act fused into A fragment
        const v2f* gRow = G2 + (size_t)((k0 >> 1) + half) * NF + l16;
        #pragma unroll
        for (int t = 0; t < 4; ++t) {
            v2f Bv = gRow[nB + t * 16];
            acc[t] = __builtin_amdgcn_wmma_f32_16x16x4_f32(
                false, A, false, Bv, (short)0, acc[t], false, false);
        }
    }

    float s2 = 0.0f;
    #pragma unroll
    for (int t = 0; t < 4; ++t) {
        const int col = nB + t * 16 + l16;
        #pragma unroll
        for (int r = 0; r < 8; ++r) {
            const int idx = (m0 + r + 8 * half) * NF + col;
            const float uo = u_in[idx];
            const float du = excite[idx] - uo - acc[t][r];
            u_out[idx] = uo + LRATE * du;
            s2 += du * du;
        }
    }

    // wave32 reduction of ||du||^2, one atomic per wave
    #pragma unroll
    for (int off = 16; off > 0; off >>= 1) s2 += __shfl_down(s2, off);
    if (lane == 0) atomicAdd(&norms[iter], s2);

    if (blockIdx.x == 0 && threadIdx.x == 0) *cur = out_is_b;  // record live buffer
}

// ---------------------------------------------------------------------------
// final: out = threshold(u_current)
// ---------------------------------------------------------------------------
__global__ void lca_final_kernel(const float* __restrict__ uA,
                                 const float* __restrict__ uB,
                                 const int* __restrict__ cur,
                                 float* __restrict__ out) {
    const int idx = blockIdx.x * blockDim.x + threadIdx.x;
    const float* u = (*cur) ? uB : uA;
    if (idx < BATCH * NF) out[idx] = thr(u[idx]);
}

extern "C" void kernel_launch(void* const* d_in, const int* in_sizes, int n_in,
                              void* d_out, int out_size, void* d_ws, size_t ws_size,
                              hipStream_t stream) {
    const float* images  = (const float*)d_in[0];   // [8192, 64, 64] fp32
    const float* filters = (const float*)d_in[1];   // [64, 64, 256] fp32, pre-normalized in setup

    // Workspace layout (~29.6 MB): uA | uB | excite | phiPack | Gpack | norms | cur
    float* uA      = (float*)d_ws;
    float* uB      = uA + (size_t)BATCH * NF;
    float* exciteB = uB + (size_t)BATCH * NF;
    float* phiPack = exciteB + (size_t)BATCH * NF;
    float* Gpack   = phiPack + (size_t)WH * NF;
    float* norms   = Gpack + (size_t)NF * NF;
    int*   cur     = (int*)(norms + MAXITER + 8);

    (void)in_sizes; (void)n_in; (void)out_size; (void)ws_size;

    lca_init_kernel<<<(BATCH * NF + 255) / 256, 256, 0, stream>>>(uA, norms, cur);
    lca_pack_phi_kernel<<<(WH * NF + 255) / 256, 256, 0, stream>>>(filters, phiPack);
    lca_gram_kernel<<<32, 256, 0, stream>>>(phiPack, Gpack);
    lca_excite_kernel<<<256, 256, 0, stream>>>(images, phiPack, exciteB);

    for (int i = 0; i < MAXITER; ++i) {
        const float* uin  = (i & 1) ? uB : uA;
        float*       uout = (i & 1) ? uA : uB;
        const int out_is_b = (i & 1) ? 0 : 1;
        lca_iter_kernel<<<256, 256, 0, stream>>>(uin, uout, exciteB, Gpack, norms, cur, i, out_is_b);
    }

    lca_final_kernel<<<(BATCH * NF + 255) / 256, 256, 0, stream>>>(uA, uB, cur, (float*)d_out);
}